// Wave_Conv_36850819399877
// MI455X (gfx1250) — compile-verified
//
#include <hip/hip_runtime.h>
#include <hip/hip_bf16.h>

typedef __attribute__((ext_vector_type(2))) float v2f;
typedef __attribute__((ext_vector_type(8))) float v8f;
typedef __attribute__((ext_vector_type(4))) int   v4i;

#define KEEP_THRE 1e-4f

// ---------------- async global->LDS path (gfx1250), with safe fallback ----------------
#if defined(__has_builtin)
#if __has_builtin(__builtin_amdgcn_global_load_async_to_lds_b128)
#define ASYNC_LDS 1
#warning "gfx1250: using __builtin_amdgcn_global_load_async_to_lds_b128 (ASYNCcnt path)"
#endif
#endif
#ifndef ASYNC_LDS
#define ASYNC_LDS 0
#warning "gfx1250: async-to-LDS builtin not found; using VGPR-staging fallback"
#endif

__device__ __forceinline__ void copy16_g2l(const float* __restrict__ g, float* l) {
#if ASYNC_LDS
    // signature (from clang diagnostic): (v4i AS1* src, v4i AS3* dst, imm offset, imm cpol)
    __builtin_amdgcn_global_load_async_to_lds_b128(
        (__attribute__((address_space(1))) v4i*)(g),
        (__attribute__((address_space(3))) v4i*)(l),
        0, 0);
#else
    const float4 v = *(const float4*)g;
    l[0] = v.x; l[1] = v.y; l[2] = v.z; l[3] = v.w;
#endif
}

__device__ __forceinline__ void wait_async_lds() {
#if ASYNC_LDS
#if __has_builtin(__builtin_amdgcn_s_wait_asynccnt)
    __builtin_amdgcn_s_wait_asynccnt(0);
#else
    asm volatile("s_wait_asynccnt 0x0" ::: "memory");
#endif
#endif
}

enum GemmMode { MODE_NN = 0, MODE_TN = 1, MODE_NT = 2 };

// Block tile 128(M) x 64(N), 128 threads = 4 waves, each wave a 64x32 region
// (4x2 WMMA tiles, v_wmma_f32_16x16x4_f32). Double-buffered LDS, one barrier/iter.
// MODE_NN: C = A@B            A[M,K], B[K,N] row-major
// MODE_TN: C = A^T@B          A[K,M], B[K,N] row-major
// MODE_NT: C[m,n] = sum_k A[m,k]*scl[k]*B[n,k]   A[M,K], B[N,K] row-major
template <int MODE, bool BIAS, bool RELU>
__global__ __launch_bounds__(128)
void wmma_gemm_kernel(const float* __restrict__ A, int lda,
                      const float* __restrict__ B, int ldb,
                      const float* __restrict__ scl, int sclStride,
                      const float* __restrict__ bias,
                      float* __restrict__ C, int ldc, int K)
{
    constexpr int BM = 128, BN = 64, KT = 16;
    constexpr int PK  = 20;   // pitch, k-contiguous tiles  (80B rows: 16B-aligned, conflict-free b64 reads)
    constexpr int PBN = 72;   // pitch, n-contiguous B tile (288B rows: 16B-aligned)
    constexpr int PBM = 136;  // pitch, m-contiguous A tile for TN (544B rows: 16B-aligned)

    constexpr int ASZ = (MODE == MODE_TN) ? (KT * PBM) : (BM * PK);
    constexpr int BSZ = (MODE == MODE_NT) ? (BN * PK) : (KT * PBN);
    __shared__ float lsA[2 * ASZ];
    __shared__ float lsB[2 * BSZ];

    const int tid   = threadIdx.x;
    const int wave  = tid >> 5;      // 0..3 (wave32)
    const int lane  = tid & 31;
    const int lhalf = lane >> 4;     // 0/1
    const int lmod  = lane & 15;

    const int bm = blockIdx.y * BM;
    const int bn = blockIdx.x * BN;
    const int wm = (wave >> 1) * 64; // wave row offset inside block
    const int wn = (wave & 1) * 32;  // wave col offset inside block

    v8f acc[4][2] = {};

    auto loadTiles = [&](int k0, int buf) {
        float* dA = lsA + buf * ASZ;
        float* dB = lsB + buf * BSZ;
        if constexpr (MODE == MODE_TN) {
            // A[K,M]: tile 16(k) x 128(m) -> dA[16][PBM]
            #pragma unroll
            for (int q = 0; q < 4; ++q) {
                int idx = q * 128 + tid;          // 512 float4 slots
                int r = idx >> 5, c4 = idx & 31;
                copy16_g2l(A + (size_t)(k0 + r) * lda + bm + c4 * 4, &dA[r * PBM + c4 * 4]);
            }
        } else {
            // A[M,K]: tile 128(m) x 16(k) -> dA[128][PK]
            #pragma unroll
            for (int q = 0; q < 4; ++q) {
                int idx = q * 128 + tid;
                int r = idx >> 2, c4 = idx & 3;
                copy16_g2l(A + (size_t)(bm + r) * lda + k0 + c4 * 4, &dA[r * PK + c4 * 4]);
            }
        }
        if constexpr (MODE == MODE_NT) {
            // B[N,K]: tile 64(n) x 16(k), scaled by scl[k] -> dB[64][PK] (needs VGPR path)
            #pragma unroll
            for (int q = 0; q < 2; ++q) {
                int idx = q * 128 + tid;          // 256 float4 slots
                int r = idx >> 2, c4 = idx & 3;
                float4 v = *(const float4*)(B + (size_t)(bn + r) * ldb + k0 + c4 * 4);
                int kb = k0 + c4 * 4;
                v.x *= scl[(size_t)(kb + 0) * sclStride];
                v.y *= scl[(size_t)(kb + 1) * sclStride];
                v.z *= scl[(size_t)(kb + 2) * sclStride];
                v.w *= scl[(size_t)(kb + 3) * sclStride];
                float* d = &dB[r * PK + c4 * 4];
                d[0] = v.x; d[1] = v.y; d[2] = v.z; d[3] = v.w;
            }
        } else {
            // B[K,N]: tile 16(k) x 64(n) -> dB[16][PBN]
            #pragma unroll
            for (int q = 0; q < 2; ++q) {
                int idx = q * 128 + tid;
                int r = idx >> 4, c4 = idx & 15;
                copy16_g2l(B + (size_t)(k0 + r) * ldb + bn + c4 * 4, &dB[r * PBN + c4 * 4]);
            }
        }
    };

    const int nsteps = K / KT;
    loadTiles(0, 0);

    for (int s = 0; s < nsteps; ++s) {
        const int cur = s & 1, nxt = cur ^ 1;
        wait_async_lds();        // my async writes into buf[cur] have landed
        __syncthreads();         // everyone's tile visible; buf[nxt] consumers (iter s-1) done
        if (s + 1 < nsteps) loadTiles((s + 1) * KT, nxt);

        const float* tA = lsA + cur * ASZ;
        const float* tB = lsB + cur * BSZ;
        #pragma unroll
        for (int kk = 0; kk < KT; kk += 4) {
            const int kfrag = kk + 2 * lhalf;    // K index of fragment element v=0
            v2f a[4], b[2];
            #pragma unroll
            for (int i = 0; i < 4; ++i) {
                if constexpr (MODE == MODE_TN) {
                    int col = wm + i * 16 + lmod;
                    a[i][0] = tA[(kfrag + 0) * PBM + col];
                    a[i][1] = tA[(kfrag + 1) * PBM + col];
                } else {
                    int row = wm + i * 16 + lmod;
                    a[i][0] = tA[row * PK + kfrag + 0];
                    a[i][1] = tA[row * PK + kfrag + 1];
                }
            }
            #pragma unroll
            for (int j = 0; j < 2; ++j) {
                if constexpr (MODE == MODE_NT) {
                    int row = wn + j * 16 + lmod;
                    b[j][0] = tB[row * PK + kfrag + 0];
                    b[j][1] = tB[row * PK + kfrag + 1];
                } else {
                    int col = wn + j * 16 + lmod;
                    b[j][0] = tB[(kfrag + 0) * PBN + col];
                    b[j][1] = tB[(kfrag + 1) * PBN + col];
                }
            }
            #pragma unroll
            for (int i = 0; i < 4; ++i)
                #pragma unroll
                for (int j = 0; j < 2; ++j)
                    acc[i][j] = __builtin_amdgcn_wmma_f32_16x16x4_f32(
                        false, a[i], false, b[j], (short)0, acc[i][j], false, false);
        }
    }

    // ---------------- epilogue ----------------
    #pragma unroll
    for (int i = 0; i < 4; ++i) {
        #pragma unroll
        for (int j = 0; j < 2; ++j) {
            const int n  = bn + wn + j * 16 + lmod;
            const float bv = BIAS ? bias[n] : 0.0f;
            #pragma unroll
            for (int v = 0; v < 8; ++v) {
                const int m = bm + wm + i * 16 + 8 * lhalf + v;
                float val = acc[i][j][v] + bv;
                if (RELU) val = fmaxf(val, 0.0f);
                C[(size_t)m * ldc + n] = val;
            }
        }
    }
}

// ---------- deterministic two-stage column sum-of-squares reduction ----------
__global__ void colnorm_partial_kernel(const float* __restrict__ F, float* __restrict__ partial,
                                       int rows, int cols, int chunks)
{
    int j = blockIdx.x * blockDim.x + threadIdx.x;
    if (j >= cols) return;
    int span = rows / chunks;
    int r0 = blockIdx.y * span, r1 = r0 + span;
    float s = 0.0f;
    for (int i = r0; i < r1; ++i) {
        float v = F[(size_t)i * cols + j];
        s += v * v;
    }
    partial[(size_t)blockIdx.y * cols + j] = s;
}

__global__ void colnorm_finish_kernel(const float* __restrict__ partial, float* __restrict__ norms,
                                      int cols, int chunks)
{
    int j = blockIdx.x * blockDim.x + threadIdx.x;
    if (j >= cols) return;
    float s = 0.0f;
    for (int c = 0; c < chunks; ++c) s += partial[(size_t)c * cols + j];
    norms[j] = sqrtf(s);
}

// Fm = where(|F/n| > thre, F/n, 0) in place
__global__ void normalize_mask_kernel(float* __restrict__ F,
                                      const float* __restrict__ norms,
                                      int rows, int cols)
{
    int idx = blockIdx.x * blockDim.x + threadIdx.x;
    if (idx >= rows * cols) return;
    int j = idx & (cols - 1);              // cols == 2048 (pow2)
    float n = fmaxf(norms[j], 1e-12f);
    float v = F[idx] / n;
    F[idx] = (fabsf(v) > KEEP_THRE) ? v : 0.0f;
}

extern "C" void kernel_launch(void* const* d_in, const int* in_sizes, int n_in,
                              void* d_out, int out_size, void* d_ws, size_t ws_size,
                              hipStream_t stream) {
    (void)in_sizes; (void)n_in; (void)out_size; (void)ws_size;

    constexpr int NNODES = 2048;
    constexpr int DH     = 768;
    constexpr int T      = 4;
    constexpr int SO     = 192;   // DH / T
    constexpr int CHUNKS = 32;

    const float* x   = (const float*)d_in[0];   // [NNODES, DH]
    const float* evc = (const float*)d_in[1];   // [NNODES, NNODES]
    const float* fsa = (const float*)d_in[2];   // [1, NNODES, T]
    const float* Wg  = (const float*)d_in[3];   // [T, DH, SO]
    const float* bg  = (const float*)d_in[4];   // [T, SO]
    const float* fW  = (const float*)d_in[5];   // [DH, DH]
    const float* fb  = (const float*)d_in[6];   // [DH]
    float* out = (float*)d_out;                 // [NNODES, DH]

    // workspace carve-up (~26.5 MB of floats)
    float* F     = (float*)d_ws;                         // NNODES*NNODES
    float* XW    = F     + (size_t)NNODES * NNODES;      // NNODES*SO
    float* H1    = XW    + (size_t)NNODES * SO;          // NNODES*SO
    float* comb  = H1    + (size_t)NNODES * SO;          // NNODES*DH
    float* norms = comb  + (size_t)NNODES * DH;          // NNODES
    float* part  = norms + NNODES;                       // CHUNKS*NNODES

    const dim3 blk(128);

    for (int t = 0; t < T; ++t) {
        // XW = x @ W_gcn[t]                  [2048,768]x[768,192]
        wmma_gemm_kernel<MODE_NN, false, false><<<dim3(SO / 64, NNODES / 128), blk, 0, stream>>>(
            x, DH, Wg + (size_t)t * DH * SO, SO, nullptr, 0, nullptr, XW, SO, DH);

        // F = evc @ (diag(s_t) @ evc^T)      [2048,2048]x[2048,2048], s_t folded into B-tile load
        wmma_gemm_kernel<MODE_NT, false, false><<<dim3(NNODES / 64, NNODES / 128), blk, 0, stream>>>(
            evc, NNODES, evc, NNODES, fsa + t, T, nullptr, F, NNODES, NNODES);

        // column norms (two-stage deterministic), then normalize + threshold-mask in place
        colnorm_partial_kernel<<<dim3(NNODES / 256, CHUNKS), dim3(256), 0, stream>>>(
            F, part, NNODES, NNODES, CHUNKS);
        colnorm_finish_kernel<<<dim3(NNODES / 256), dim3(256), 0, stream>>>(
            part, norms, NNODES, CHUNKS);
        normalize_mask_kernel<<<dim3((NNODES * NNODES) / 256), dim3(256), 0, stream>>>(
            F, norms, NNODES, NNODES);

        // H1 = relu(Fm^T @ XW + b_gcn[t])    [2048,2048]^T x [2048,192]
        wmma_gemm_kernel<MODE_TN, true, true><<<dim3(SO / 64, NNODES / 128), blk, 0, stream>>>(
            F, NNODES, XW, SO, nullptr, 0, bg + (size_t)t * SO, H1, SO, NNODES);

        // comb[:, t*SO : (t+1)*SO] = Fm^T @ H1
        wmma_gemm_kernel<MODE_TN, false, false><<<dim3(SO / 64, NNODES / 128), blk, 0, stream>>>(
            F, NNODES, H1, SO, nullptr, 0, nullptr, comb + (size_t)t * SO, DH, NNODES);
    }

    // out = relu(comb @ fusion_W + fusion_b) [2048,768]x[768,768]
    wmma_gemm_kernel<MODE_NN, true, true><<<dim3(DH / 64, NNODES / 128), blk, 0, stream>>>(
        comb, DH, fW, DH, nullptr, 0, fb, out, DH, DH);
}